// Mask_30683246362706
// MI455X (gfx1250) — compile-verified
//
#include <hip/hip_runtime.h>
#include <stdint.h>

// Problem constants (match reference)
#define ROWS  4096
#define COLS  8192
#define KSEL  4096          // top-k per row
#define TPB   512           // 16 waves (wave32)
#define VPT   (COLS / TPB)  // 16 elements per thread
#define NWAVE (TPB / 32)    // 16

// ---- CDNA5 async global<->LDS path (ASYNCcnt) -------------------------------

__device__ __forceinline__ unsigned ldsAddr(const void* p) {
  // Low 32 bits of a flat shared pointer are the LDS byte address (ISA 10.2).
  return (unsigned)(uintptr_t)p;
}

__device__ __forceinline__ void asyncLoadB128_NT(const void* gbase,
                                                 unsigned voff, unsigned lds) {
  // GVS mode: mem = SGPR64 + VGPR32; VDST vgpr = LDS byte address. Streaming.
  asm volatile("global_load_async_to_lds_b128 %0, %1, %2 th:TH_LOAD_NT"
               :: "v"(lds), "v"(voff), "s"(gbase) : "memory");
}

__device__ __forceinline__ void asyncStoreB128_NT(const void* gbase,
                                                  unsigned voff, unsigned lds) {
  asm volatile("global_store_async_from_lds_b128 %0, %1, %2 th:TH_STORE_NT"
               :: "v"(voff), "v"(lds), "s"(gbase) : "memory");
}

__device__ __forceinline__ void waitAsync0() {
  asm volatile("s_wait_asynccnt 0" ::: "memory");
}

// ---- helpers ---------------------------------------------------------------

// Monotonic float -> uint transform (ascending order as unsigned).
__device__ __forceinline__ unsigned f2ord(float f) {
  unsigned u = __float_as_uint(f);
  return u ^ (unsigned)(((int)u >> 31) | 0x80000000);
}
__device__ __forceinline__ float ord2f(unsigned k) {
  unsigned u = k ^ (((int)k < 0) ? 0x80000000u : 0xFFFFFFFFu);
  return __uint_as_float(u);
}
__device__ __forceinline__ float sigmoid_stable(float x) {
  float e = __expf(-fabsf(x));
  float r = 1.0f / (1.0f + e);
  return (x >= 0.0f) ? r : (1.0f - r);
}

// ---- kernel: one workgroup per row -----------------------------------------

__global__ __launch_bounds__(TPB) void gumbel_topk_mask(
    const float* __restrict__ zg, const float* __restrict__ eg,
    float* __restrict__ og) {
  __shared__ __align__(16) unsigned keybuf[COLS];   // eps -> keys -> output (32 KB)
  __shared__ unsigned whist[NWAVE][256];            // per-wave histograms (16 KB)
  __shared__ unsigned sfx[256];                     // suffix sums
  __shared__ unsigned scanbuf[TPB];                 // tie prefix scan
  __shared__ unsigned bcast[2];                     // {digit, new_rem}

  const int tid = threadIdx.x;
  const size_t rowoff = (size_t)blockIdx.x * COLS;
  const float* zr = zg + rowoff;
  const float* er = eg + rowoff;
  float* outr = og + rowoff;
  const unsigned lds0 = ldsAddr(&keybuf[0]);

  // 1) Issue eps row -> LDS async DMA (ASYNCcnt, NT) first, then the z row
  //    -> VGPRs in one asm block that waits LOADcnt internally. Both memory
  //    streams are in flight together; single join before the barrier.
  #pragma unroll
  for (int j = 0; j < 4; ++j) {
    unsigned boff = (unsigned)(j * 8192 + tid * 16);
    asyncLoadB128_NT(er, boff, lds0 + boff);
  }

  uint4 zb[4];
  {
    unsigned o0 = (unsigned)(tid * 16);
    unsigned o1 = o0 + 8192u;
    unsigned o2 = o0 + 16384u;
    unsigned o3 = o0 + 24576u;
    asm volatile(
        "global_load_b128 %0, %4, %8 th:TH_LOAD_NT\n\t"
        "global_load_b128 %1, %5, %8 th:TH_LOAD_NT\n\t"
        "global_load_b128 %2, %6, %8 th:TH_LOAD_NT\n\t"
        "global_load_b128 %3, %7, %8 th:TH_LOAD_NT\n\t"
        "s_wait_loadcnt 0"
        : "=&v"(zb[0]), "=&v"(zb[1]), "=&v"(zb[2]), "=&v"(zb[3])
        : "v"(o0), "v"(o1), "v"(o2), "v"(o3), "s"(zr)
        : "memory");
  }
  waitAsync0();
  __syncthreads();

  // 2) keys = orderable bits of s = z - log(-log(eps)), written in place.
  #pragma unroll
  for (int j = 0; j < 4; ++j) {
    int base = j * 2048 + tid * 4;
    uint4 ev = *reinterpret_cast<uint4*>(&keybuf[base]);
    float s0 = __uint_as_float(zb[j].x) - __logf(-__logf(__uint_as_float(ev.x)));
    float s1 = __uint_as_float(zb[j].y) - __logf(-__logf(__uint_as_float(ev.y)));
    float s2 = __uint_as_float(zb[j].z) - __logf(-__logf(__uint_as_float(ev.z)));
    float s3 = __uint_as_float(zb[j].w) - __logf(-__logf(__uint_as_float(ev.w)));
    uint4 kv;
    kv.x = f2ord(s0); kv.y = f2ord(s1); kv.z = f2ord(s2); kv.w = f2ord(s3);
    *reinterpret_cast<uint4*>(&keybuf[base]) = kv;
  }
  __syncthreads();

  // 3) radix select: find K-th largest key (MSB-first, 8 bits/pass).
  unsigned pref = 0, hiMask = 0, rem = KSEL;
  const int wv = tid >> 5;
  for (int shift = 24; shift >= 0; shift -= 8) {
    #pragma unroll
    for (int k = 0; k < (NWAVE * 256) / TPB; ++k)
      (&whist[0][0])[tid + k * TPB] = 0;
    __syncthreads();

    #pragma unroll
    for (int j = 0; j < VPT; ++j) {
      unsigned key = keybuf[tid + j * TPB];
      if ((key & hiMask) == pref)
        atomicAdd(&whist[wv][(key >> shift) & 255], 1u);
    }
    __syncthreads();

    if (tid < 256) {
      unsigned c = 0;
      #pragma unroll
      for (int w = 0; w < NWAVE; ++w) c += whist[w][tid];
      sfx[tid] = c;
    }
    __syncthreads();

    // suffix sums: sfx[d] = count of candidates with digit >= d
    for (int off = 1; off < 256; off <<= 1) {
      unsigned v = 0;
      if (tid < 256) v = sfx[tid] + ((tid + off < 256) ? sfx[tid + off] : 0u);
      __syncthreads();
      if (tid < 256) sfx[tid] = v;
      __syncthreads();
    }

    if (tid < 256) {
      unsigned gt = (tid == 255) ? 0u : sfx[tid + 1];
      if (sfx[tid] >= rem && gt < rem) {
        bcast[0] = (unsigned)tid;
        bcast[1] = rem - gt;
      }
    }
    __syncthreads();
    pref   |= bcast[0] << shift;
    hiMask |= 0xFFu << shift;
    rem     = bcast[1];
    __syncthreads();
  }
  const unsigned thr = pref;    // K-th largest key
  const unsigned quota = rem;   // how many ties (lowest index first) to accept

  // 4) exact tie-break by ascending index + straight-through output.
  const int cbase = tid * VPT;  // contiguous chunk keeps global index order
  unsigned kk[VPT];
  unsigned cnt = 0;
  #pragma unroll
  for (int j = 0; j < VPT; ++j) {
    kk[j] = keybuf[cbase + j];
    cnt += (kk[j] == thr) ? 1u : 0u;
  }
  scanbuf[tid] = cnt;
  __syncthreads();
  for (int off = 1; off < TPB; off <<= 1) {
    unsigned v = scanbuf[tid] + ((tid >= off) ? scanbuf[tid - off] : 0u);
    __syncthreads();
    scanbuf[tid] = v;
    __syncthreads();
  }
  unsigned tiePos = scanbuf[tid] - cnt;  // exclusive prefix of ties before chunk

  #pragma unroll
  for (int j = 0; j < VPT; ++j) {
    unsigned key = kk[j];
    float hard;
    if (key > thr) {
      hard = 1.0f;
    } else if (key == thr && tiePos < quota) {
      hard = 1.0f;
      ++tiePos;
    } else {
      hard = 0.0f;
    }
    float gm = sigmoid_stable(1.5f * ord2f(key));     // sigmoid(s / (2/3))
    keybuf[cbase + j] = __float_as_uint((hard - gm) + gm);  // ST forward
  }
  __syncthreads();

  // 5) async DMA the finished row LDS -> global (streaming store).
  #pragma unroll
  for (int j = 0; j < 4; ++j) {
    unsigned boff = (unsigned)(j * 8192 + tid * 16);
    asyncStoreB128_NT(outr, boff, lds0 + boff);
  }
  waitAsync0();
}

// ---- harness entry ---------------------------------------------------------

extern "C" void kernel_launch(void* const* d_in, const int* in_sizes, int n_in,
                              void* d_out, int out_size, void* d_ws,
                              size_t ws_size, hipStream_t stream) {
  (void)in_sizes; (void)n_in; (void)out_size; (void)d_ws; (void)ws_size;
  const float* z   = (const float*)d_in[0];  // z_loga  [ROWS*COLS] f32
  const float* eps = (const float*)d_in[1];  // eps     [ROWS*COLS] f32
  float* out = (float*)d_out;                // mask    [ROWS*COLS] f32
  gumbel_topk_mask<<<ROWS, TPB, 0, stream>>>(z, eps, out);
}